// Ising_sampling2_52974126628895
// MI455X (gfx1250) — compile-verified
//
#include <hip/hip_runtime.h>

#define BATCH 262144
#define DIM 64
#define WAVES 4
#define ROWS_PER_WAVE 32
#define XSTR 68   // padded row stride (floats) for x tile: banks (4r+k)%64, conflict-free
#define ASTR 20   // padded row stride (floats) for alpha staging

typedef __attribute__((ext_vector_type(2))) float v2f;
typedef __attribute__((ext_vector_type(8))) float v8f;

__device__ __forceinline__ void ds_fence() {
  __asm__ volatile("s_wait_dscnt 0" ::: "memory");
}

__global__ __launch_bounds__(WAVES * 32) void ising_sweep_kernel(
    const float* __restrict__ Q, const float* __restrict__ delta,
    const float* __restrict__ x0, const float* __restrict__ u,
    const int* __restrict__ nlayers_p, const int* __restrict__ const_p,
    float* __restrict__ out_z, float* __restrict__ out_e)
{
  __shared__ float Msh[DIM * DIM];
  __shared__ float dsh[DIM];
  __shared__ float xsh[WAVES][ROWS_PER_WAVE * XSTR];
  __shared__ float ash[WAVES][ROWS_PER_WAVE * ASTR];

  const int tid    = threadIdx.x;
  const int wid    = tid >> 5;
  const int lane   = tid & 31;
  const int laneLo = lane & 15;
  const int laneHi = lane >> 4;                       // 0 or 1
  const int rowBase = (blockIdx.x * WAVES + wid) * ROWS_PER_WAVE;

  // Build symmetric M in LDS from packed strict-lower-triangle Q (row-major tril).
  for (int idx = tid; idx < DIM * DIM; idx += blockDim.x) {
    int r = idx >> 6, c = idx & 63;
    float v = 0.0f;
    if (r > c)      v = Q[(r * (r - 1)) / 2 + c];
    else if (c > r) v = Q[(c * (c - 1)) / 2 + r];
    Msh[idx] = v;
  }
  if (tid < DIM) dsh[tid] = delta[tid];

  // Load this wave's 32x64 x tile into padded LDS (coalesced global reads).
  float* xt = xsh[wid];
  for (int i = lane; i < ROWS_PER_WAVE * DIM; i += 32) {
    int r = i >> 6, c = i & 63;
    xt[r * XSTR + c] = x0[(size_t)(rowBase + r) * DIM + c];
  }
  __syncthreads();

  const int   nL  = nlayers_p[0];
  const float cst = (float)const_p[0];
  float* at = ash[wid];

  for (int l = 0; l < nL; ++l) {
    for (int t = 0; t < 4; ++t) {
      // ---- WMMA GEMM: alpha(32x16) = X(32x64) * M[:, 16t .. 16t+15] ----
      v8f acc0 = {}; v8f acc1 = {};
      #pragma unroll
      for (int s = 0; s < 16; ++s) {
        const int k0 = 4 * s;
        const int ka = k0 + 2 * laneHi;
        v2f a0 = *(const v2f*)&xt[laneLo * XSTR + ka];
        v2f a1 = *(const v2f*)&xt[(16 + laneLo) * XSTR + ka];
        v2f b;
        b.x = Msh[(ka + 0) * DIM + 16 * t + laneLo];
        b.y = Msh[(ka + 1) * DIM + 16 * t + laneLo];
        acc0 = __builtin_amdgcn_wmma_f32_16x16x4_f32(false, a0, false, b,
                                                     (short)0, acc0, false, false);
        acc1 = __builtin_amdgcn_wmma_f32_16x16x4_f32(false, a1, false, b,
                                                     (short)0, acc1, false, false);
      }
      const float dk = dsh[16 * t + laneLo];
      // Stage C tiles to LDS to transpose into "lane = batch row" layout.
      #pragma unroll
      for (int v = 0; v < 8; ++v) {
        at[(8 * laneHi + v) * ASTR + laneLo]        = acc0[v] + dk;
        at[(16 + 8 * laneHi + v) * ASTR + laneLo]   = acc1[v] + dk;
      }
      ds_fence();

      float alpha[16];
      #pragma unroll
      for (int n = 0; n < 16; ++n) alpha[n] = at[lane * ASTR + n];
      float xr[16];
      #pragma unroll
      for (int n = 0; n < 16; ++n) xr[n] = xt[lane * XSTR + 16 * t + n];

      const unsigned long long ubase =
          (((unsigned long long)l * BATCH + (unsigned long long)(rowBase + lane)) << 6)
          + (unsigned)(16 * t);

      // ---- sequential Gauss-Seidel within the 16-wide block ----
      #pragma unroll
      for (int kk = 0; kk < 16; ++kk) {
        const int k = 16 * t + kk;
        float a    = alpha[kk];
        float uval = u[ubase + kk];
        float sig  = 1.0f / (1.0f + __expf(-2.0f * a));      // sigmoid(2a)
        float y    = cst * (sig - uval);
        float nv   = 2.0f / (1.0f + __expf(-2.0f * y)) - 1.0f; // tanh(y)
        float dx   = nv - xr[kk];
        xr[kk] = nv;
        #pragma unroll
        for (int n = kk + 1; n < 16; ++n)
          alpha[n] += Msh[k * DIM + 16 * t + n] * dx;          // uniform LDS broadcast
      }
      #pragma unroll
      for (int n = 0; n < 16; ++n) xt[lane * XSTR + 16 * t + n] = xr[n];
      ds_fence();
    }
  }

  // ---- energy: e[r] = sum_k (0.5*h[k] + delta[k]) * z[r][k],  h = z @ M ----
  float e = 0.0f;
  for (int t = 0; t < 4; ++t) {
    v8f acc0 = {}; v8f acc1 = {};
    #pragma unroll
    for (int s = 0; s < 16; ++s) {
      const int k0 = 4 * s;
      const int ka = k0 + 2 * laneHi;
      v2f a0 = *(const v2f*)&xt[laneLo * XSTR + ka];
      v2f a1 = *(const v2f*)&xt[(16 + laneLo) * XSTR + ka];
      v2f b;
      b.x = Msh[(ka + 0) * DIM + 16 * t + laneLo];
      b.y = Msh[(ka + 1) * DIM + 16 * t + laneLo];
      acc0 = __builtin_amdgcn_wmma_f32_16x16x4_f32(false, a0, false, b,
                                                   (short)0, acc0, false, false);
      acc1 = __builtin_amdgcn_wmma_f32_16x16x4_f32(false, a1, false, b,
                                                   (short)0, acc1, false, false);
    }
    #pragma unroll
    for (int v = 0; v < 8; ++v) {
      at[(8 * laneHi + v) * ASTR + laneLo]      = acc0[v];
      at[(16 + 8 * laneHi + v) * ASTR + laneLo] = acc1[v];
    }
    ds_fence();
    #pragma unroll
    for (int n = 0; n < 16; ++n) {
      float z = xt[lane * XSTR + 16 * t + n];
      e += (0.5f * at[lane * ASTR + n] + dsh[16 * t + n]) * z;
    }
    ds_fence();   // at[] reused next t
  }

  // Write z (coalesced) and energy.
  for (int i = lane; i < ROWS_PER_WAVE * DIM; i += 32) {
    int r = i >> 6, c = i & 63;
    out_z[(size_t)(rowBase + r) * DIM + c] = xt[r * XSTR + c];
  }
  out_e[rowBase + lane] = e;
}

extern "C" void kernel_launch(void* const* d_in, const int* in_sizes, int n_in,
                              void* d_out, int out_size, void* d_ws, size_t ws_size,
                              hipStream_t stream) {
  // setup_inputs order: inputs(B,1) [unused], Q(2016), delta(64), x0(B*64),
  //                     u(L*B*64), n_layers(1), const(1)
  const float* Q     = (const float*)d_in[1];
  const float* delta = (const float*)d_in[2];
  const float* x0    = (const float*)d_in[3];
  const float* u     = (const float*)d_in[4];
  const int*   nL    = (const int*)d_in[5];
  const int*   cst   = (const int*)d_in[6];

  float* out_z = (float*)d_out;                    // (B, 64)
  float* out_e = out_z + (size_t)BATCH * DIM;      // (B,)

  dim3 grid(BATCH / (WAVES * ROWS_PER_WAVE));      // 2048 workgroups
  ising_sweep_kernel<<<grid, WAVES * 32, 0, stream>>>(Q, delta, x0, u, nL, cst,
                                                      out_z, out_e);
}